// KeypointsLoss_65730179498694
// MI455X (gfx1250) — compile-verified
//
#include <hip/hip_runtime.h>

typedef float v2f __attribute__((ext_vector_type(2)));
typedef float v4f __attribute__((ext_vector_type(4)));
typedef float v8f __attribute__((ext_vector_type(8)));

#define HW   128   // H == W (reference requires this)
#define KPTS 17
#define NB   64

// Exact f32 wave32 reduction via V_WMMA_F32_16X16X4_F32.
// A(16x4): VGPR0 = lane acc (lanes 0-15 -> K=0, lanes 16-31 -> K=2), VGPR1 = 0.
// B(4x16) = ones. C[i][j] = acc[i] + acc[i+16] for i=0..15.
// Per-lane sum of the 8 C VGPRs: lanes 0-15 hold r0+..+r7, lanes 16-31 hold r8+..+r15.
// Wave total = s(lane0) + s(lane16).
__device__ __forceinline__ float wave_fold_wmma(float acc) {
    v2f a; a[0] = acc;  a[1] = 0.0f;
    v2f b; b[0] = 1.0f; b[1] = 1.0f;
    v8f c = {};
    c = __builtin_amdgcn_wmma_f32_16x16x4_f32(false, a, false, b, (short)0, c, false, false);
    return ((c[0] + c[1]) + (c[2] + c[3])) + ((c[4] + c[5]) + (c[6] + c[7]));
}

// One block per (b,k) slice: stream 128x128 f32 tile, accumulate (pred - target)^2.
// target = valid ? exp(-(h-x)^2/18) * exp(-(w-y)^2/18) : 0  (separable Gaussian).
__global__ __launch_bounds__(256) void KeypointsLoss_se_partial(
    const float* __restrict__ pred, const float* __restrict__ kp,
    const int* __restrict__ vis, float* __restrict__ partial)
{
    const int bk   = blockIdx.x;        // b*KPTS + k
    const int tid  = threadIdx.x;       // 0..255
    const int lane = tid & 31;          // column group: float4 at w = 4*lane
    const int wrow = tid >> 5;          // 0..7 row within row-group

    // Keypoint -> integer grid position (truncation, matching astype(int32)).
    const float kx = kp[2 * bk + 0];
    const float ky = kp[2 * bk + 1];
    const int x = (int)(kx * (float)(HW - 1));   // compared against h axis
    const int y = (int)(ky * (float)(HW - 1));   // compared against w axis
    const bool valid = (vis[bk] > 0) & (x >= 0) & (x < HW) & (y >= 0) & (y < HW);

    // Per-lane column factors (each thread owns the same 4 columns all rows).
    const float inv18 = 1.0f / 18.0f;
    const float yf = (float)y;
    const float w0 = (float)(4 * lane);
    float d;
    d = w0          - yf;  const float ey0 = __expf(-d * d * inv18);
    d = w0 + 1.0f   - yf;  const float ey1 = __expf(-d * d * inv18);
    d = w0 + 2.0f   - yf;  const float ey2 = __expf(-d * d * inv18);
    d = w0 + 3.0f   - yf;  const float ey3 = __expf(-d * d * inv18);

    const v4f* __restrict__ src = (const v4f*)pred + (size_t)bk * (HW * HW / 4);

    float acc = 0.0f;
    #pragma unroll
    for (int it = 0; it < 16; ++it) {            // 16 row-groups x 8 rows = 128 rows
        const int row = it * 8 + wrow;
        const v4f p = __builtin_nontemporal_load(&src[row * (HW / 4) + lane]);
        const float dr = (float)(row - x);
        const float ax = valid ? __expf(-dr * dr * inv18) : 0.0f;  // row factor (0 kills target)
        float d0 = p[0] - ax * ey0;  acc = fmaf(d0, d0, acc);
        float d1 = p[1] - ax * ey1;  acc = fmaf(d1, d1, acc);
        float d2 = p[2] - ax * ey2;  acc = fmaf(d2, d2, acc);
        float d3 = p[3] - ax * ey3;  acc = fmaf(d3, d3, acc);
    }

    // Wave fold via WMMA, then tiny LDS combine of 8 waves x 2 half-sums.
    const float s = wave_fold_wmma(acc);
    __shared__ float red[16];
    if ((lane & 15) == 0) red[(tid >> 5) * 2 + (lane >> 4)] = s;
    __syncthreads();
    if (tid == 0) {
        float t = 0.0f;
        #pragma unroll
        for (int i = 0; i < 16; ++i) t += red[i];
        partial[bk] = t;
    }
}

// One wave: group 1088 partials by sample, normalize, mean.
__global__ __launch_bounds__(32) void KeypointsLoss_finalize(
    const float* __restrict__ partial, const int* __restrict__ vis,
    float* __restrict__ out)
{
    const int lane = threadIdx.x;    // 32 lanes, 2 samples each
    float v = 0.0f;
    #pragma unroll
    for (int rep = 0; rep < 2; ++rep) {
        const int b = lane + rep * 32;
        float se = 0.0f;
        int dv = 0;
        #pragma unroll
        for (int k = 0; k < KPTS; ++k) {
            se += partial[b * KPTS + k];
            dv += vis[b * KPTS + k];
        }
        v += se / ((float)dv + 1e-6f);
    }
    const float s = wave_fold_wmma(v);
    __shared__ float r2[2];
    if ((lane & 15) == 0) r2[lane >> 4] = s;
    __syncthreads();
    if (lane == 0) out[0] = (r2[0] + r2[1]) * (1.0f / (float)NB);
}

extern "C" void kernel_launch(void* const* d_in, const int* in_sizes, int n_in,
                              void* d_out, int out_size, void* d_ws, size_t ws_size,
                              hipStream_t stream) {
    const float* pred = (const float*)d_in[0];   // [B,K,H,W] f32
    const float* kp   = (const float*)d_in[1];   // [B,K,2]   f32
    const int*   vis  = (const int*)d_in[2];     // [B,K]     i32
    float* out = (float*)d_out;
    float* partial = (float*)d_ws;               // BK floats of scratch

    const int BK = in_sizes[2];                  // B*K = 1088
    KeypointsLoss_se_partial<<<BK, 256, 0, stream>>>(pred, kp, vis, partial);
    KeypointsLoss_finalize<<<1, 32, 0, stream>>>(partial, vis, out);
}